// RetrievalDotLoss_26946624815702
// MI455X (gfx1250) — compile-verified
//
#include <hip/hip_runtime.h>
#include <hip/hip_bf16.h>

typedef __attribute__((ext_vector_type(2))) float v2f;
typedef __attribute__((ext_vector_type(8))) float v8f;

// Rowwise dot product as diag(Q · Doc^T) using V_WMMA_F32_16X16X4_F32.
// Each wave32 owns 16 rows; 64 WMMA steps over K=256.
// Fragment layout (32-bit A 16x4 / B 4x16, ISA 7.12.2):
//   lane L<16  : row = L,     K = {k0,   k0+1}
//   lane L>=16 : row = L-16,  K = {k0+2, k0+3}
// Identical addressing for A (from Q) and B (from Doc, since B[k][n] = Doc[n][k]).
__global__ __launch_bounds__(256)
void RetrievalDotLoss_wmma_kernel(const float* __restrict__ q,
                                  const float* __restrict__ doc,
                                  float* __restrict__ out) {
    constexpr int D = 256;
    const int lane    = threadIdx.x & 31;
    const int wave    = threadIdx.x >> 5;
    const int rowBase = blockIdx.x * 128 + wave * 16;   // 16 rows per wave
    const int r       = lane & 15;                      // row within tile
    const int kh      = (lane >> 4) << 1;               // K sub-offset: 0 or 2

    const size_t base = (size_t)(rowBase + r) * D + kh;
    const float* __restrict__ qp = q   + base;
    const float* __restrict__ dp = doc + base;

    v8f c = {0.f, 0.f, 0.f, 0.f, 0.f, 0.f, 0.f, 0.f};

    #pragma unroll 4
    for (int k0 = 0; k0 < D; k0 += 4) {
        if ((k0 & 63) == 0) {
            // stream-ahead hints (speculative, dropped if OOB)
            __builtin_prefetch(qp + k0 + 256, 0, 0);
            __builtin_prefetch(dp + k0 + 256, 0, 0);
        }
        v2f a = __builtin_nontemporal_load((const v2f*)(qp + k0));
        v2f b = __builtin_nontemporal_load((const v2f*)(dp + k0));
        // D = A x B + C  (f32, exact — same math as scalar FMA chain)
        c = __builtin_amdgcn_wmma_f32_16x16x4_f32(
                /*neg_a=*/false, a, /*neg_b=*/false, b,
                /*c_mod=*/(short)0, c, /*reuse_a=*/false, /*reuse_b=*/false);
    }

    // Diagonal extraction from 16x16 f32 C layout:
    //   i < 8 : C vgpr i,   lane i
    //   i >= 8: C vgpr i-8, lane i+16   (lanes 24..31)
    const bool lo = (lane < 8);
    const bool hi = (lane >= 24);
    const int  sel = lo ? lane : (lane - 24);   // 0..7 where active
    float v = c[0];
    v = (sel == 1) ? c[1] : v;
    v = (sel == 2) ? c[2] : v;
    v = (sel == 3) ? c[3] : v;
    v = (sel == 4) ? c[4] : v;
    v = (sel == 5) ? c[5] : v;
    v = (sel == 6) ? c[6] : v;
    v = (sel == 7) ? c[7] : v;

    if (lo || hi) {
        const int row = rowBase + (lo ? lane : (lane - 16));
        __builtin_nontemporal_store(-v, out + row);
    }
}

extern "C" void kernel_launch(void* const* d_in, const int* in_sizes, int n_in,
                              void* d_out, int out_size, void* d_ws, size_t ws_size,
                              hipStream_t stream) {
    const float* q   = (const float*)d_in[0];
    const float* doc = (const float*)d_in[1];
    float* out = (float*)d_out;

    const int D = 256;
    const int n = in_sizes[0] / D;          // 65536 rows
    const int blocks = n / 128;             // 128 rows per 256-thread block (8 waves x 16 rows)
    RetrievalDotLoss_wmma_kernel<<<blocks, 256, 0, stream>>>(q, doc, out);
}